// ComplexWindowAttention_14018773254208
// MI455X (gfx1250) — compile-verified
//
#include <hip/hip_runtime.h>

typedef __attribute__((ext_vector_type(2))) float v2f;
typedef __attribute__((ext_vector_type(8))) float v8f;

#define B_WIN 1024
#define NTOK  64
#define CCH   256
#define NH    8
#define HD    32
#define QSCALE 0.17677669529663687f   // 32^-0.5

// LDS row pitches (chosen for 8B alignment + conflict-free bank access)
#define PQK 34   // Q/K planes: even (b64-aligned), 34 mod 64 = 34 -> distinct bank pairs
#define PS  67   // S planes: 67 mod 64 = 3 -> conflict-free b32 + softmax rows
#define PV  68   // V^T planes: even, 68 mod 64 = 4 -> halves on disjoint mod-4 residues

static __device__ __forceinline__ v8f wmma4(v2f a, v2f b, v8f c) {
  // D = A(16x4) x B(4x16) + C, fp32 in/out
  return __builtin_amdgcn_wmma_f32_16x16x4_f32(false, a, false, b, (short)0, c,
                                               false, false);
}

// ---------------------------------------------------------------------------
// Kernel 1: fused QKV projection (this head's 64x96 slice) + complex window
// attention for one (window b, head h). One workgroup = 256 threads = 8 waves.
// ---------------------------------------------------------------------------
__global__ __launch_bounds__(256) void cwa_qkv_attn(
    const float* __restrict__ x,     // [B, N, C] complex interleaved
    const float* __restrict__ wqkv,  // [3C, C]  complex
    const float* __restrict__ bqkv,  // [3C]     complex
    const float* __restrict__ btab,  // [225, H] complex
    const int*   __restrict__ ridx,  // [64, 64] int32
    float* __restrict__ y)           // [B*N, C] complex interleaved (workspace)
{
  __shared__ float ldsA[4 * 64 * PQK];  // Qr,Qi,Kr,Ki  ->  Sr,Si (64*PS each)
  __shared__ float ldsV[2 * 32 * PV];   // Vt re, Vt im  ([HD][NTOK])

  float* Qr  = ldsA;
  float* Qi  = ldsA + 64 * PQK;
  float* Kr  = ldsA + 2 * 64 * PQK;
  float* Ki  = ldsA + 3 * 64 * PQK;
  float* Vtr = ldsV;
  float* Vti = ldsV + 32 * PV;
  float* Sr  = ldsA;                    // aliases Q/K region (after barrier)
  float* Si  = ldsA + 64 * PS;          // 2*64*67 = 8576 <= 4*64*34 = 8704

  const int b    = blockIdx.x / NH;
  const int h    = blockIdx.x % NH;
  const int tid  = threadIdx.x;
  const int wave = tid >> 5;
  const int lane = tid & 31;
  const int ln   = lane & 15;          // tile column / A-row index
  const int khalf  = lane >> 4;        // 0: K 0..1, 1: K 2..3
  const int rowAdd = khalf * 8;        // C/D row offset for upper half-wave

  const float* xb = x + (size_t)b * NTOK * CCH * 2;

  // ---- Phase 1: QKV slice. Each wave: one m-tile x three j-tiles (A reuse).
  {
    const int mt = wave & 3;
    const int jh = wave >> 2;          // 0 -> j-tiles 0..2, 1 -> 3..5
    const int i0 = mt * 16;

    int jb[3];
    const float4* wrow[3];
#pragma unroll
    for (int q = 0; q < 3; ++q) {
      const int jt = jh * 3 + q;       // 0..5: s = jt>>1, d-half = jt&1
      jb[q] = (jt >> 1) * CCH + h * HD + (jt & 1) * 16;
      wrow[q] = (const float4*)(wqkv + (size_t)(jb[q] + ln) * CCH * 2);
    }
    const float4* xrow = (const float4*)(xb + (size_t)(i0 + ln) * CCH * 2);

    v8f arr[3] = {}, aii[3] = {}, ari[3] = {}, air[3] = {};
#pragma unroll 2
    for (int k0 = 0; k0 < CCH; k0 += 4) {
      const int kk = (k0 >> 1) + khalf;         // float4 = 2 complex elements
      const float4 xa = xrow[kk];
      const v2f Ar = {xa.x, xa.z}, Ai = {xa.y, xa.w};
#pragma unroll
      for (int q = 0; q < 3; ++q) {
        const float4 wb = wrow[q][kk];
        const v2f Br = {wb.x, wb.z}, Bi = {wb.y, wb.w};
        arr[q] = wmma4(Ar, Br, arr[q]);
        aii[q] = wmma4(Ai, Bi, aii[q]);
        ari[q] = wmma4(Ar, Bi, ari[q]);
        air[q] = wmma4(Ai, Br, air[q]);
      }
    }
#pragma unroll
    for (int q = 0; q < 3; ++q) {
      const int jt = jh * 3 + q;
      const int s  = jt >> 1;
      const int d0 = (jt & 1) * 16;
      const float2 bj = ((const float2*)bqkv)[jb[q] + ln];
      const int col = d0 + ln;
      for (int r = 0; r < 8; ++r) {
        const int row = i0 + r + rowAdd;
        const float ore = arr[q][r] - aii[q][r] + bj.x;
        const float oim = ari[q][r] + air[q][r] + bj.y;
        if (s == 0) {                           // Q (pre-scaled)
          Qr[row * PQK + col] = ore * QSCALE;
          Qi[row * PQK + col] = oim * QSCALE;
        } else if (s == 1) {                    // K
          Kr[row * PQK + col] = ore;
          Ki[row * PQK + col] = oim;
        } else {                                // V stored transposed [d][tok]
          Vtr[col * PV + row] = ore;
          Vti[col * PV + row] = oim;
        }
      }
    }
  }
  __syncthreads();

  // ---- Phase 2: S = Q K^T (complex, 64x64). Each wave: 2 tiles, shared Q.
  v8f srr[2] = {}, sii[2] = {}, sri[2] = {}, sir[2] = {};
  const int si0 = (wave & 3) * 16;
  const int sj0 = (wave >> 2) * 16;    // second tile at sj0 + 32
  {
    const float* qr  = Qr + (si0 + ln) * PQK;
    const float* qi  = Qi + (si0 + ln) * PQK;
    const float* k0r = Kr + (sj0 + ln) * PQK;
    const float* k0i = Ki + (sj0 + ln) * PQK;
    const float* k1r = k0r + 32 * PQK;
    const float* k1i = k0i + 32 * PQK;
#pragma unroll
    for (int kx = 0; kx < HD; kx += 4) {
      const int kk = kx + khalf * 2;           // even -> 8B-aligned b64 reads
      const v2f Ar = *(const v2f*)(qr + kk);
      const v2f Ai = *(const v2f*)(qi + kk);
      const v2f B0r = *(const v2f*)(k0r + kk);
      const v2f B0i = *(const v2f*)(k0i + kk);
      const v2f B1r = *(const v2f*)(k1r + kk);
      const v2f B1i = *(const v2f*)(k1i + kk);
      srr[0] = wmma4(Ar, B0r, srr[0]);
      sii[0] = wmma4(Ai, B0i, sii[0]);
      sri[0] = wmma4(Ar, B0i, sri[0]);
      sir[0] = wmma4(Ai, B0r, sir[0]);
      srr[1] = wmma4(Ar, B1r, srr[1]);
      sii[1] = wmma4(Ai, B1i, sii[1]);
      sri[1] = wmma4(Ar, B1i, sri[1]);
      sir[1] = wmma4(Ai, B1r, sir[1]);
    }
  }
  __syncthreads();   // all Q/K reads done before S overwrites that LDS region

  // store S + relative-position bias into (aliased) Sr/Si planes
#pragma unroll
  for (int u = 0; u < 2; ++u) {
    const int j = sj0 + u * 32 + ln;
    for (int r = 0; r < 8; ++r) {
      const int i  = si0 + r + rowAdd;
      const int bi = ridx[i * 64 + j];
      const float2 bv = ((const float2*)btab)[bi * NH + h];
      Sr[i * PS + j] = srr[u][r] - sii[u][r] + bv.x;
      Si[i * PS + j] = sri[u][r] + sir[u][r] + bv.y;
    }
  }
  __syncthreads();

  // ---- Phase 3: split softmax over last axis, independently on re & im.
  if (tid < 128) {
    float* row = ((tid < 64) ? Sr : Si) + (tid & 63) * PS;
    float mx = row[0];
    for (int j = 1; j < 64; ++j) mx = fmaxf(mx, row[j]);
    float sum = 0.0f;
    for (int j = 0; j < 64; ++j) { float e = __expf(row[j] - mx); row[j] = e; sum += e; }
    const float inv = 1.0f / sum;
    for (int j = 0; j < 64; ++j) row[j] *= inv;
  }
  __syncthreads();

  // ---- Phase 4: O = A V (complex). 8 tiles (4 m x 2 d), 1 per wave.
  {
    const int i0 = (wave & 3) * 16;
    const int d0 = (wave >> 2) * 16;
    v8f arr = {}, aii = {}, ari = {}, air = {};
    const float* sr = Sr  + (i0 + ln) * PS;
    const float* si = Si  + (i0 + ln) * PS;
    const float* vr = Vtr + (d0 + ln) * PV;
    const float* vi = Vti + (d0 + ln) * PV;
#pragma unroll 4
    for (int kx = 0; kx < NTOK; kx += 4) {
      const int kk = kx + khalf * 2;
      const v2f Ar = {sr[kk], sr[kk + 1]};
      const v2f Ai = {si[kk], si[kk + 1]};
      const v2f Br = *(const v2f*)(vr + kk);   // (d0+ln)*68 + kk even -> b64
      const v2f Bi = *(const v2f*)(vi + kk);
      arr = wmma4(Ar, Br, arr);
      aii = wmma4(Ai, Bi, aii);
      ari = wmma4(Ar, Bi, ari);
      air = wmma4(Ai, Br, air);
    }
    const int c = h * HD + d0 + ln;
    for (int r = 0; r < 8; ++r) {
      const int i = i0 + r + rowAdd;
      float2 o;
      o.x = arr[r] - aii[r];
      o.y = ari[r] + air[r];
      ((float2*)y)[((size_t)b * NTOK + i) * CCH + c] = o;
    }
  }
}

// ---------------------------------------------------------------------------
// Kernel 2: out = y @ w_proj^T + b_proj (complex GEMM, M=65536 N=256 K=256).
// One wave per 16x64 output strip (4 j-tiles, shared A operand).
// ---------------------------------------------------------------------------
__global__ __launch_bounds__(256) void cwa_proj(
    const float* __restrict__ yy,    // [B*N, C] complex
    const float* __restrict__ wp,    // [C, C]   complex
    const float* __restrict__ bp,    // [C]      complex
    float* __restrict__ out)         // [B*N, C] complex
{
  const int tid  = threadIdx.x;
  const int wave = tid >> 5;
  const int lane = tid & 31;
  const int ln   = lane & 15;
  const int khalf  = lane >> 4;
  const int rowAdd = khalf * 8;

  const int wid   = blockIdx.x * 8 + wave;   // 16384 strips
  const int jq    = wid & 3;                 // 4 j-quads of 64 columns
  const int mtile = wid >> 2;                // 4096 m-tiles

  const float4* yrow = (const float4*)(yy + (size_t)(mtile * 16 + ln) * CCH * 2);
  const float4* wrow[4];
#pragma unroll
  for (int q = 0; q < 4; ++q)
    wrow[q] = (const float4*)(wp + (size_t)(jq * 64 + q * 16 + ln) * CCH * 2);

  v8f arr[4] = {}, aii[4] = {}, ari[4] = {}, air[4] = {};
#pragma unroll 2
  for (int k0 = 0; k0 < CCH; k0 += 4) {
    const int kk = (k0 >> 1) + khalf;
    const float4 ya = yrow[kk];
    const v2f Ar = {ya.x, ya.z}, Ai = {ya.y, ya.w};
#pragma unroll
    for (int q = 0; q < 4; ++q) {
      const float4 wb = wrow[q][kk];
      const v2f Br = {wb.x, wb.z}, Bi = {wb.y, wb.w};
      arr[q] = wmma4(Ar, Br, arr[q]);
      aii[q] = wmma4(Ai, Bi, aii[q]);
      ari[q] = wmma4(Ar, Bi, ari[q]);
      air[q] = wmma4(Ai, Br, air[q]);
    }
  }
#pragma unroll
  for (int q = 0; q < 4; ++q) {
    const int jcol = jq * 64 + q * 16 + ln;
    const float2 bj = ((const float2*)bp)[jcol];
    for (int r = 0; r < 8; ++r) {
      const int row = mtile * 16 + r + rowAdd;
      float2 o;
      o.x = arr[q][r] - aii[q][r] + bj.x;
      o.y = ari[q][r] + air[q][r] + bj.y;
      ((float2*)out)[(size_t)row * CCH + jcol] = o;
    }
  }
}

extern "C" void kernel_launch(void* const* d_in, const int* in_sizes, int n_in,
                              void* d_out, int out_size, void* d_ws, size_t ws_size,
                              hipStream_t stream) {
  (void)in_sizes; (void)n_in; (void)out_size; (void)ws_size;
  const float* x    = (const float*)d_in[0];
  const float* wqkv = (const float*)d_in[1];
  const float* bqkv = (const float*)d_in[2];
  const float* wp   = (const float*)d_in[3];
  const float* bp   = (const float*)d_in[4];
  const float* btab = (const float*)d_in[5];
  const int*   ridx = (const int*)d_in[6];

  float* y = (float*)d_ws;   // [B*N, C] complex = 134.2 MB scratch

  // 1024 windows x 8 heads, 256 threads (8 wave32) each
  cwa_qkv_attn<<<B_WIN * NH, 256, 0, stream>>>(x, wqkv, bqkv, btab, ridx, y);

  // 16384 strips / 8 waves per block = 2048 blocks
  cwa_proj<<<2048, 256, 0, stream>>>(y, wp, bp, (float*)d_out);
}